// DarkChannelPrior_326417514529
// MI455X (gfx1250) — compile-verified
//
#include <hip/hip_runtime.h>
#include <stdint.h>

#define BN 4
#define CN 3
#define HN 720
#define WN 1280
#define HWN (HN*WN)
#define KR 7            // dark-channel min radius (15x15)
#define RR 40           // guided filter radius
#define TOPK 921        // int(720*1280*0.001)
#define OMEGA 0.95f
#define EPSF 0.001f

#define PAD 64          // zero padding per side for hbox LDS rows
#define WROW 1412       // 1280+128 rounded so stride % 64 banks == 4 (conflict-free), even

// vbox tiling
#define VTX 64          // columns per block
#define VTY 48          // output rows per block
#define VROWS 128       // max staged input rows (VTY + 2*RR)

typedef __attribute__((ext_vector_type(2))) float v2f;
typedef __attribute__((ext_vector_type(8))) float v8f;
typedef __attribute__((ext_vector_type(4))) unsigned tdm_g0_t;
typedef __attribute__((ext_vector_type(8))) int      tdm_g1_t;
typedef __attribute__((ext_vector_type(4))) int      tdm_g2_t;

#if __has_builtin(__builtin_amdgcn_tensor_load_to_lds) && __has_builtin(__builtin_amdgcn_s_wait_tensorcnt)
#define HAVE_TDM 1
#else
#define HAVE_TDM 0
#endif

__device__ __forceinline__ int refl(int i, int n) {
  if (i < 0) i = -i;
  if (i >= n) i = 2*n - 2 - i;
  return i;
}

// ---------------- ctl layout (uint32 words) ----------------
// [0..3] hi bin  [4..7] count>hi-bin  [8..11] threshold bits
// [12..23] airlight bits  [24..35] img min bits  [36..47] img max bits

__global__ void dcp_init_ctl(unsigned* ctl) {
  int t = threadIdx.x;
  if (t < 24) ctl[t] = 0u;
  else if (t < 36) ctl[t] = 0x7F800000u;
  else if (t < 48) ctl[t] = 0u;
}

__global__ void dcp_zero_u32(unsigned* p, int n) {
  for (int i = blockIdx.x*blockDim.x + threadIdx.x; i < n; i += gridDim.x*blockDim.x)
    p[i] = 0u;
}

__global__ void dcp_minmax(const float* __restrict__ img, unsigned* ctl) {
  int p = blockIdx.y;                          // b*3+c
  const float* src = img + (size_t)p * HWN;
  unsigned mn = 0x7F800000u, mx = 0u;
  for (int i = blockIdx.x*blockDim.x + threadIdx.x; i < HWN; i += gridDim.x*blockDim.x) {
    unsigned v = __float_as_uint(src[i]);
    mn = v < mn ? v : mn;
    mx = v > mx ? v : mx;
  }
  __shared__ unsigned smn[256], smx[256];
  smn[threadIdx.x] = mn; smx[threadIdx.x] = mx;
  __syncthreads();
  for (int s = 128; s > 0; s >>= 1) {
    if (threadIdx.x < s) {
      smn[threadIdx.x] = smn[threadIdx.x] < smn[threadIdx.x+s] ? smn[threadIdx.x] : smn[threadIdx.x+s];
      smx[threadIdx.x] = smx[threadIdx.x] > smx[threadIdx.x+s] ? smx[threadIdx.x] : smx[threadIdx.x+s];
    }
    __syncthreads();
  }
  if (threadIdx.x == 0) {
    atomicMin(&ctl[24 + p], smn[0]);
    atomicMax(&ctl[36 + p], smx[0]);
  }
}

__global__ void dcp_cmin(const float* __restrict__ img, float* __restrict__ out,
                         const float* __restrict__ air, int divAir) {
  int i = blockIdx.x*blockDim.x + threadIdx.x;
  if (i >= BN*HWN) return;
  int b = i / HWN, px = i - b*HWN;
  const float* base = img + (size_t)b*3*HWN + px;
  float v0 = base[0], v1 = base[HWN], v2 = base[2*HWN];
  if (divAir) { v0 /= air[b*3+0]; v1 /= air[b*3+1]; v2 /= air[b*3+2]; }
  out[i] = fminf(v0, fminf(v1, v2));
}

__global__ void dcp_hmin15(const float* __restrict__ in, float* __restrict__ out) {
  int i = blockIdx.x*blockDim.x + threadIdx.x;
  if (i >= BN*HWN) return;
  int b = i / HWN, px = i - b*HWN;
  int y = px / WN, x = px - y*WN;
  const float* row = in + (size_t)b*HWN + (size_t)y*WN;
  float m = 1e30f;
  #pragma unroll
  for (int d = -KR; d <= KR; ++d) m = fminf(m, row[refl(x + d, WN)]);
  out[i] = m;
}

__global__ void dcp_vmin15(const float* __restrict__ in, float* __restrict__ out, int mode) {
  int i = blockIdx.x*blockDim.x + threadIdx.x;
  if (i >= BN*HWN) return;
  int b = i / HWN, px = i - b*HWN;
  int y = px / WN, x = px - y*WN;
  const float* pl = in + (size_t)b*HWN;
  float m = 1e30f;
  #pragma unroll
  for (int d = -KR; d <= KR; ++d) m = fminf(m, pl[(size_t)refl(y + d, HN)*WN + x]);
  out[i] = (mode == 0) ? m : fmaxf(0.2f, 1.0f - OMEGA * m);
}

__global__ void dcp_hist_hi(const float* __restrict__ dc, unsigned* hist) {
  int i = blockIdx.x*blockDim.x + threadIdx.x;
  if (i >= BN*HWN) return;
  int b = i / HWN;
  unsigned bits = __float_as_uint(dc[i]);
  atomicAdd(&hist[(size_t)b*65536 + (bits >> 16)], 1u);
}

__global__ void dcp_find_hi(const unsigned* hist, unsigned* ctl) {
  if (threadIdx.x != 0) return;
  int b = blockIdx.x;
  const unsigned* hb = hist + (size_t)b*65536;
  unsigned cum = 0; int bin = 65535;
  for (; bin >= 0; --bin) {
    unsigned c = hb[bin];
    if (cum + c >= TOPK) break;
    cum += c;
  }
  if (bin < 0) bin = 0;
  ctl[b] = (unsigned)bin;
  ctl[4 + b] = cum;
}

__global__ void dcp_hist_lo(const float* __restrict__ dc, unsigned* hist, const unsigned* ctl) {
  int i = blockIdx.x*blockDim.x + threadIdx.x;
  if (i >= BN*HWN) return;
  int b = i / HWN;
  unsigned bits = __float_as_uint(dc[i]);
  if ((bits >> 16) == ctl[b])
    atomicAdd(&hist[(size_t)b*65536 + (bits & 0xFFFFu)], 1u);
}

__global__ void dcp_find_lo(const unsigned* hist, unsigned* ctl) {
  if (threadIdx.x != 0) return;
  int b = blockIdx.x;
  const unsigned* hb = hist + (size_t)b*65536;
  unsigned cum = ctl[4 + b]; int bin = 65535;
  for (; bin >= 0; --bin) {
    cum += hb[bin];
    if (cum >= TOPK) break;
  }
  if (bin < 0) bin = 0;
  ctl[8 + b] = (ctl[b] << 16) | (unsigned)bin;
}

__global__ void dcp_air(const float* __restrict__ dc, const float* __restrict__ img, unsigned* ctl) {
  int i = blockIdx.x*blockDim.x + threadIdx.x;
  if (i >= BN*HWN) return;
  int b = i / HWN, px = i - b*HWN;
  float thr = __uint_as_float(ctl[8 + b]);
  if (dc[i] >= thr) {
    const float* base = img + (size_t)b*3*HWN + px;
    atomicMax(&ctl[12 + b*3 + 0], __float_as_uint(base[0]));
    atomicMax(&ctl[12 + b*3 + 1], __float_as_uint(base[HWN]));
    atomicMax(&ctl[12 + b*3 + 2], __float_as_uint(base[2*HWN]));
  }
}

__global__ void dcp_air_fin(const unsigned* ctl, float* air_out) {
  int t = threadIdx.x;
  if (t < 12) air_out[t] = fminf(__uint_as_float(ctl[12 + t]), 220.0f);
}

// -----------------------------------------------------------------------
// Horizontal box-sum (r=40, clipped == zero-padded conv) via WMMA.
// LDS rows zero-padded by 64 each side -> every access unconditional.
// Row stride 1412 floats (== 4 mod 64 banks): conflict-free fragment loads.
// 16 x v_wmma_f32_16x16x4_f32 (all-ones B) cover the 64 interior taps;
// exactly 17 boundary taps per element, branchless:
//   u(j) = (x-40)+j + (j+N>=16 ? 64 : 0)   covers [x-40,x0-25] u [x0+40,x+40]
// -----------------------------------------------------------------------
__global__ void __launch_bounds__(256) dcp_hbox(const float* __restrict__ s1, int c1,
                                                const float* __restrict__ s2, int c2, int useS2,
                                                const unsigned* __restrict__ ctl, int b,
                                                float* __restrict__ out) {
  __shared__ float rows[16][WROW];     // ~88 KB of the 320 KB WGP LDS
  int y0 = blockIdx.y * 16;

  float mn1 = 0.f, iv1 = 1.f, mn2 = 0.f, iv2 = 1.f;
  if (c1 >= 0) {
    mn1 = __uint_as_float(ctl[24 + b*3 + c1]);
    iv1 = 1.f / (__uint_as_float(ctl[36 + b*3 + c1]) - mn1);
  }
  if (useS2 && c2 >= 0) {
    mn2 = __uint_as_float(ctl[24 + b*3 + c2]);
    iv2 = 1.f / (__uint_as_float(ctl[36 + b*3 + c2]) - mn2);
  }
  for (int r = 0; r < 16; ++r) {
    const float* src1 = s1 + (size_t)(y0 + r) * WN;
    const float* src2 = useS2 ? (s2 + (size_t)(y0 + r) * WN) : src1;
    __builtin_prefetch(src1 + WN, 0, 1);
    for (int xp = threadIdx.x; xp < WN + 2*PAD; xp += 256) {
      int x = xp - PAD;
      float v = 0.f;
      if (x >= 0 && x < WN) {
        v = src1[x];
        if (c1 >= 0) v = (v - mn1) * iv1;
        if (useS2) {
          float w = src2[x];
          if (c2 >= 0) w = (w - mn2) * iv2;
          v *= w;
        }
      }
      rows[r][xp] = v;
    }
  }
  __syncthreads();

  int lane = threadIdx.x & 31;
  int wv   = threadIdx.x >> 5;               // wave32
  int x0   = (blockIdx.x * 8 + wv) * 16;

  v8f acc = {0.f, 0.f, 0.f, 0.f, 0.f, 0.f, 0.f, 0.f};
  v2f ones; ones.x = 1.f; ones.y = 1.f;      // all-ones B: layout-independent
  int r   = lane & 15;                        // A row (M)
  int sub = (lane >> 4) << 1;                 // K sub-position, 16x4 f32 A layout

  const float* rp = &rows[r][PAD];            // u==0 at image col 0; always in-bounds
  #pragma unroll
  for (int t = 0; t < 16; ++t) {
    int u = x0 - 24 + 4*t + sub;              // even -> 8B-aligned ds_load_b64
    v2f a = *(const v2f*)(rp + u);
    acc = __builtin_amdgcn_wmma_f32_16x16x4_f32(false, a, false, ones,
                                                (short)0, acc, false, false);
  }

  int N   = lane & 15;
  int mhi = (lane >> 4) << 3;                 // lanes 16-31 hold M+8
  int x   = x0 + N;

  // 17 boundary tap offsets, lane-dependent only (shared by all 8 C elements)
  int uu[17];
  #pragma unroll
  for (int j = 0; j < 17; ++j)
    uu[j] = (x - RR) + j + ((j + N >= 16) ? 64 : 0);

  #pragma unroll
  for (int v = 0; v < 8; ++v) {
    int M = v + mhi;
    const float* mp = &rows[M][PAD];
    float s = acc[v];
    #pragma unroll
    for (int j = 0; j < 17; ++j) s += mp[uu[j]];
    out[(size_t)(y0 + M) * WN + x] = s;
  }
}

// -----------------------------------------------------------------------
// Vertical box-sum with analytic /N.  64x48 output tile per block; input
// column tile staged to LDS by the Tensor Data Mover (TDM), consumed after
// s_wait_tensorcnt + barrier.  Running column sum: ~2 LDS ops per output.
// -----------------------------------------------------------------------
#if HAVE_TDM
__device__ __forceinline__ void tdm_load_2d(unsigned lds_off, const float* gsrc,
                                            int tile_w, int tile_h, int pitch_elems) {
  unsigned long long ga = (unsigned long long)(size_t)gsrc;
  tdm_g0_t g0;
  g0[0] = 1u;                                            // count=1, user mode
  g0[1] = lds_off;                                       // LDS byte address
  g0[2] = (unsigned)(ga & 0xFFFFFFFFull);                // global_addr[31:0]
  g0[3] = (unsigned)((ga >> 32) & 0x01FFFFFFull) | (2u << 30);  // [56:32] | type=2
  unsigned td0 = (unsigned)tile_w;                       // tensor_dim0 (tile never OOB)
  unsigned td1 = (unsigned)tile_h;                       // tensor_dim1
  unsigned long long st0 = (unsigned long long)pitch_elems;
  tdm_g1_t g1;
  g1[0] = (int)(2u << 16);                               // data_size = 4 bytes
  g1[1] = (int)((td0 & 0xFFFFu) << 16);                  // tensor_dim0[15:0]
  g1[2] = (int)(((td0 >> 16) & 0xFFFFu) | ((td1 & 0xFFFFu) << 16));
  g1[3] = (int)(((td1 >> 16) & 0xFFFFu) | ((unsigned)tile_w << 16));  // tile_dim0
  g1[4] = (int)((unsigned)tile_h & 0xFFFFu);             // tile_dim1 (tile_dim2=0)
  g1[5] = (int)(st0 & 0xFFFFFFFFull);                    // tensor_dim0_stride[31:0]
  g1[6] = (int)((st0 >> 32) & 0xFFFFull);                // stride[47:32] (dim1_stride=0)
  g1[7] = 0;
  tdm_g2_t z4 = {0, 0, 0, 0};
  tdm_g1_t z8 = {0, 0, 0, 0, 0, 0, 0, 0};
  __builtin_amdgcn_tensor_load_to_lds(g0, g1, z4, z4, z8, 0);
}
#endif

__global__ void __launch_bounds__(256) dcp_vbox(const float* __restrict__ in,
                                                float* __restrict__ out) {
  __shared__ float tile[VROWS][VTX];          // 32 KB
  int x0 = blockIdx.x * VTX;
  int y0 = blockIdx.y * VTY;
  int r0 = max(y0 - RR, 0);
  int r1 = min(y0 + VTY - 1 + RR, HN - 1);
  int nrows = r1 - r0 + 1;

#if HAVE_TDM
  if (threadIdx.x < 32) {
    tdm_load_2d((unsigned)(size_t)&tile[0][0],
                in + (size_t)r0 * WN + x0, VTX, nrows, WN);
    __builtin_amdgcn_s_wait_tensorcnt(0);
  }
  __syncthreads();
#else
  for (int i = threadIdx.x; i < nrows * VTX; i += 256) {
    int r = i >> 6, xx = i & (VTX - 1);
    tile[r][xx] = in[(size_t)(r0 + r) * WN + x0 + xx];
  }
  __syncthreads();
#endif

  int tx = threadIdx.x & (VTX - 1);
  int ty = threadIdx.x >> 6;                  // 0..3, each covers 12 output rows
  int x  = x0 + tx;
  float invnx = 1.f / (float)(min(x + RR, WN - 1) - max(x - RR, 0) + 1);

  int ybase = y0 + ty * (VTY/4);
  int lo = max(ybase - RR, 0), hi = min(ybase + RR, HN - 1);
  float s = 0.f;
  for (int r = lo; r <= hi; ++r) s += tile[r - r0][tx];
  out[(size_t)ybase * WN + x] = s * invnx / (float)(hi - lo + 1);
  #pragma unroll
  for (int k = 1; k < VTY/4; ++k) {
    int y = ybase + k;
    int nlo = max(y - RR, 0), nhi = min(y + RR, HN - 1);
    if (nhi > hi) s += tile[nhi - r0][tx];
    if (nlo > lo) s -= tile[lo - r0][tx];
    lo = nlo; hi = nhi;
    out[(size_t)y * WN + x] = s * invnx / (float)(hi - lo + 1);
  }
}

// per-pixel symmetric 3x3 solve (cofactors) -> a0..a2, b
__global__ void dcp_solve(const float* __restrict__ m, float* __restrict__ ab) {
  int i = blockIdx.x*blockDim.x + threadIdx.x;
  if (i >= HWN) return;
  float mI0 = m[i], mI1 = m[HWN + i], mI2 = m[2*HWN + i], mP = m[3*HWN + i];
  float c0 = m[4*HWN + i] - mI0*mP;
  float c1 = m[5*HWN + i] - mI1*mP;
  float c2 = m[6*HWN + i] - mI2*mP;
  float v00 = m[7*HWN + i]  - mI0*mI0 + EPSF;
  float v01 = m[8*HWN + i]  - mI0*mI1;
  float v02 = m[9*HWN + i]  - mI0*mI2;
  float v11 = m[10*HWN + i] - mI1*mI1 + EPSF;
  float v12 = m[11*HWN + i] - mI1*mI2;
  float v22 = m[12*HWN + i] - mI2*mI2 + EPSF;
  float A00 = v11*v22 - v12*v12;
  float A01 = v02*v12 - v01*v22;
  float A02 = v01*v12 - v02*v11;
  float A11 = v00*v22 - v02*v02;
  float A12 = v01*v02 - v00*v12;
  float A22 = v00*v11 - v01*v01;
  float inv = 1.f / (v00*A00 + v01*A01 + v02*A02);
  float a0 = (A00*c0 + A01*c1 + A02*c2) * inv;
  float a1 = (A01*c0 + A11*c1 + A12*c2) * inv;
  float a2 = (A02*c0 + A12*c1 + A22*c2) * inv;
  ab[i]         = a0;
  ab[HWN + i]   = a1;
  ab[2*HWN + i] = a2;
  ab[3*HWN + i] = mP - a0*mI0 - a1*mI1 - a2*mI2;
}

// refined_t = sum(mean_a * I_norm) + mean_b ; dehaze = (img - A)/t + A
__global__ void dcp_final(const float* __restrict__ imgb, const float* __restrict__ ab,
                          const unsigned* __restrict__ ctl, int b,
                          const float* __restrict__ air,
                          float* __restrict__ dehazeb, float* __restrict__ refb) {
  int i = blockIdx.x*blockDim.x + threadIdx.x;
  if (i >= HWN) return;
  float t = ab[3*HWN + i];
  #pragma unroll
  for (int c = 0; c < 3; ++c) {
    float mn = __uint_as_float(ctl[24 + b*3 + c]);
    float iv = 1.f / (__uint_as_float(ctl[36 + b*3 + c]) - mn);
    t += ab[(size_t)c*HWN + i] * ((imgb[(size_t)c*HWN + i] - mn) * iv);
  }
  refb[i] = t;
  #pragma unroll
  for (int c = 0; c < 3; ++c) {
    float A = air[b*3 + c];
    dehazeb[(size_t)c*HWN + i] = (imgb[(size_t)c*HWN + i] - A) / t + A;
  }
}

extern "C" void kernel_launch(void* const* d_in, const int* in_sizes, int n_in,
                              void* d_out, int out_size, void* d_ws, size_t ws_size,
                              hipStream_t stream) {
  const float* img = (const float*)d_in[0];
  float* out = (float*)d_out;

  float* d_dehaze = out;                                   // 4*3*HW
  float* d_dc     = d_dehaze + (size_t)BN*3*HWN;           // 4*HW
  float* d_air    = d_dc + (size_t)BN*HWN;                 // 12
  float* d_rawt   = d_air + BN*3;                          // 4*HW
  float* d_reft   = d_rawt + (size_t)BN*HWN;               // 4*HW

  float* pool  = (float*)d_ws;
  float* HBUF  = pool;                                     // 1 plane
  float* MEANS = pool + (size_t)HWN;                       // 13 planes
  float* ABUF  = pool + (size_t)14*HWN;                    // 4 planes
  float* FULLA = MEANS;                                    // alias (dark-channel stage)
  float* FULLB = MEANS + (size_t)4*HWN;
  unsigned* hist = (unsigned*)(pool + (size_t)18*HWN);     // 4*65536
  unsigned* ctl  = hist + (size_t)BN*65536;                // 48 words

  int nt = BN*HWN;
  int gb = (nt + 255) / 256;
  int pg = (HWN + 255) / 256;

  dcp_init_ctl<<<1, 64, 0, stream>>>(ctl);
  dcp_zero_u32<<<1024, 256, 0, stream>>>(hist, BN*65536);
  dcp_minmax<<<dim3(32, 12), 256, 0, stream>>>(img, ctl);

  dcp_cmin<<<gb, 256, 0, stream>>>(img, FULLA, d_air, 0);
  dcp_hmin15<<<gb, 256, 0, stream>>>(FULLA, FULLB);
  dcp_vmin15<<<gb, 256, 0, stream>>>(FULLB, d_dc, 0);

  dcp_hist_hi<<<gb, 256, 0, stream>>>(d_dc, hist);
  dcp_find_hi<<<BN, 32, 0, stream>>>(hist, ctl);
  dcp_zero_u32<<<1024, 256, 0, stream>>>(hist, BN*65536);
  dcp_hist_lo<<<gb, 256, 0, stream>>>(d_dc, hist, ctl);
  dcp_find_lo<<<BN, 32, 0, stream>>>(hist, ctl);
  dcp_air<<<gb, 256, 0, stream>>>(d_dc, img, ctl);
  dcp_air_fin<<<1, 32, 0, stream>>>(ctl, d_air);

  dcp_cmin<<<gb, 256, 0, stream>>>(img, FULLA, d_air, 1);
  dcp_hmin15<<<gb, 256, 0, stream>>>(FULLA, FULLB);
  dcp_vmin15<<<gb, 256, 0, stream>>>(FULLB, d_rawt, 1);

  dim3 hgrid(10, 45);                // 1280/(8*16) x 720/16
  dim3 vgrid(WN/VTX, HN/VTY);        // 20 x 15
  for (int b = 0; b < BN; ++b) {
    const float* Ib = img + (size_t)b*3*HWN;
    const float* I0 = Ib, *I1 = Ib + HWN, *I2 = Ib + 2*HWN;
    const float* pb = d_rawt + (size_t)b*HWN;
    struct Cfg { const float* s1; int c1; const float* s2; int c2; int u2; };
    Cfg cfgs[13] = {
      {I0, 0, nullptr, -1, 0}, {I1, 1, nullptr, -1, 0}, {I2, 2, nullptr, -1, 0},  // mean_I
      {pb, -1, nullptr, -1, 0},                                                   // mean_p
      {I0, 0, pb, -1, 1}, {I1, 1, pb, -1, 1}, {I2, 2, pb, -1, 1},                 // mean_Ip
      {I0, 0, I0, 0, 1}, {I0, 0, I1, 1, 1}, {I0, 0, I2, 2, 1},                    // mean_II
      {I1, 1, I1, 1, 1}, {I1, 1, I2, 2, 1}, {I2, 2, I2, 2, 1}
    };
    for (int i = 0; i < 13; ++i) {
      dcp_hbox<<<hgrid, 256, 0, stream>>>(cfgs[i].s1, cfgs[i].c1, cfgs[i].s2,
                                          cfgs[i].c2, cfgs[i].u2, ctl, b, HBUF);
      dcp_vbox<<<vgrid, 256, 0, stream>>>(HBUF, MEANS + (size_t)i*HWN);
    }
    dcp_solve<<<pg, 256, 0, stream>>>(MEANS, ABUF);
    for (int j = 0; j < 4; ++j) {
      dcp_hbox<<<hgrid, 256, 0, stream>>>(ABUF + (size_t)j*HWN, -1, nullptr, -1, 0, ctl, b, HBUF);
      dcp_vbox<<<vgrid, 256, 0, stream>>>(HBUF, ABUF + (size_t)j*HWN);
    }
    dcp_final<<<pg, 256, 0, stream>>>(Ib, ABUF, ctl, b, d_air,
                                      d_dehaze + (size_t)b*3*HWN,
                                      d_reft + (size_t)b*HWN);
  }
}